// RNNTJointNet_23785528886240
// MI455X (gfx1250) — compile-verified
//
#include <hip/hip_runtime.h>

// Problem constants (from reference): T=256, B=8, H1=H2=512, U=64, V=1024
#define TT 256
#define BB 8
#define UU 64
#define HH 512   // K dimension for both GEMMs
#define VV 1024  // N dimension

typedef float v2f __attribute__((ext_vector_type(2)));
typedef float v4f __attribute__((ext_vector_type(4)));
typedef float v8f __attribute__((ext_vector_type(8)));

// ---------------------------------------------------------------------------
// Phase 1: C[rows x 1024] = A[rows x 512] * W[512 x 1024] using
// V_WMMA_F32_16X16X4_F32. Each wave owns a 16x64 C tile (4 accumulators),
// each 256-thread block owns 16 rows x 512 cols (8 waves). A tile staged in
// LDS once per block; W streamed from L2 (W is 4MB, fully L2-resident).
//
// fLayout==1 : A is f with layout [T,B,H1]; logical row r = b*T + t
// fLayout==0 : A is g with layout [rows,H2] (contiguous)
// ---------------------------------------------------------------------------
__global__ __launch_bounds__(256)
void rnnt_joint_gemm(const float* __restrict__ A,
                     const float* __restrict__ Wp,   // W (hf) or W+H1*V (hg)
                     float* __restrict__ out,        // [rows x 1024]
                     int rowsPerBatch,               // T or U
                     int fLayout) {
  __shared__ float sA[16][HH + 4];  // pad stride to 516 floats -> spread banks

  const int tid  = threadIdx.x;
  const int lane = tid & 31;
  const int wave = tid >> 5;
  const int rowBase = blockIdx.x * 16;
  const int colBase = blockIdx.y * 512 + wave * 64;

  // ---- stage the 16x512 A tile into LDS (coalesced float4 loads) ----
  const int ab = rowBase / rowsPerBatch;   // batch index (f layout only)
  const int r0 = rowBase % rowsPerBatch;   // t0 within batch
  #pragma unroll
  for (int it = 0; it < 8; ++it) {
    const int lin = it * 256 + tid;        // 0..2047 float4 slots
    const int row = lin >> 7;              // 0..15
    const int c4  = (lin & 127) << 2;      // 0..508
    long src;
    if (fLayout) src = ((long)(r0 + row) * BB + ab) * HH + c4;  // f[t,b,h]
    else         src = (long)(rowBase + row) * HH + c4;         // g[r,h]
    const v4f v = *(const v4f*)(A + src);
    sA[row][c4 + 0] = v.x; sA[row][c4 + 1] = v.y;
    sA[row][c4 + 2] = v.z; sA[row][c4 + 3] = v.w;
  }
  __syncthreads();

  v8f acc[4] = {};                 // four 16x16 f32 accumulators (zero)

  const int lr = lane & 15;        // row/col within half-wave
  const int kh = (lane >> 4) << 1; // K sub-offset: 0 (lanes 0-15) or 2 (16-31)

  for (int k = 0; k < HH; k += 4) {
    // A fragment 16x4: lane<16 -> K=k,k+1 ; lane>=16 -> K=k+2,k+3 (row M=lr)
    v2f a;
    a.x = sA[lr][k + kh];
    a.y = sA[lr][k + kh + 1];
    #pragma unroll
    for (int n = 0; n < 4; ++n) {
      // B fragment 4x16: lane<16 -> rows K=k,k+1 ; lane>=16 -> K=k+2,k+3
      const int col = colBase + n * 16 + lr;
      v2f b;
      b.x = Wp[(long)(k + kh) * VV + col];
      b.y = Wp[(long)(k + kh + 1) * VV + col];
      // D = A*B + C   (8 args: neg_a, A, neg_b, B, c_mod, C, reuse_a, reuse_b)
      acc[n] = __builtin_amdgcn_wmma_f32_16x16x4_f32(
          false, a, false, b, (short)0, acc[n], false, false);
    }
  }

  // ---- store C: VGPR j holds rows (j | j+8), N striped over lanes ----
  const int mHalf = (lane >> 4) << 3;  // 0 or 8
  #pragma unroll
  for (int n = 0; n < 4; ++n) {
    const int col = colBase + n * 16 + lr;
    #pragma unroll
    for (int j = 0; j < 8; ++j) {
      out[(long)(rowBase + mHalf + j) * VV + col] = acc[n][j];
    }
  }
}

// ---------------------------------------------------------------------------
// Phase 2: out[b,t,u,v] = mask(b,t,u) ? hf[b,t,v]+hg[b,u,v]+bias[v] : 0
// Pure store-bandwidth bound (537 MB). One float4 per thread, non-temporal
// b128 stores (output is write-once, never re-read -> keep it out of L2 so
// hf/hg/bias stay resident). Index math is all power-of-two masks/shifts.
// Block 0 also appends f_lens (8 ints, raw bits) to the output tail.
// ---------------------------------------------------------------------------
__global__ __launch_bounds__(256)
void rnnt_joint_add(const float* __restrict__ hf,
                    const float* __restrict__ hg,
                    const float* __restrict__ bias,
                    const int* __restrict__ f_lens,
                    const int* __restrict__ g_lens,
                    float* __restrict__ out) {
  const long idx = (long)blockIdx.x * 256 + threadIdx.x;  // float4 index
  const int v4 = (int)(idx & (VV / 4 - 1));   // 8 bits  : v/4
  const int u  = (int)((idx >> 8) & (UU - 1)); // 6 bits
  const int t  = (int)((idx >> 14) & (TT - 1)); // 8 bits
  const int b  = (int)(idx >> 22);             // 3 bits

  v4f r = {};
  if (t < f_lens[b] && u < g_lens[b]) {
    const v4f a = ((const v4f*)hf)[(long)(b * TT + t) * (VV / 4) + v4];
    const v4f c = ((const v4f*)hg)[(long)(b * UU + u) * (VV / 4) + v4];
    const v4f e = ((const v4f*)bias)[v4];
    r = a + c + e;
  }
  __builtin_nontemporal_store(r, (v4f*)out + idx);

  // append f_lens bits after the 134,217,728 result floats
  if (blockIdx.x == 0 && threadIdx.x < BB) {
    ((int*)out)[(long)BB * TT * UU * VV + threadIdx.x] = f_lens[threadIdx.x];
  }
}

// ---------------------------------------------------------------------------
extern "C" void kernel_launch(void* const* d_in, const int* in_sizes, int n_in,
                              void* d_out, int out_size, void* d_ws, size_t ws_size,
                              hipStream_t stream) {
  const float* f      = (const float*)d_in[0];  // [T,B,H1]
  const int*   f_lens = (const int*)  d_in[1];  // [B]
  const float* g      = (const float*)d_in[2];  // [B,U,H2]
  const int*   g_lens = (const int*)  d_in[3];  // [B]
  const float* W      = (const float*)d_in[4];  // [H1+H2, V]
  const float* bias   = (const float*)d_in[5];  // [V]
  float* out = (float*)d_out;

  float* hf = (float*)d_ws;                       // [B*T, V] = 8 MB
  float* hg = hf + (long)BB * TT * VV;            // [B*U, V] = 2 MB

  // GEMM1: hf = fb @ W[:H1]   rows = B*T = 2048 -> 128 row-tiles x 2 col-tiles
  {
    dim3 grid(BB * TT / 16, VV / 512);
    rnnt_joint_gemm<<<grid, 256, 0, stream>>>(f, W, hf, TT, 1);
  }
  // GEMM2: hg = g @ W[H1:]    rows = B*U = 512 -> 32 row-tiles x 2 col-tiles
  {
    dim3 grid(BB * UU / 16, VV / 512);
    rnnt_joint_gemm<<<grid, 256, 0, stream>>>(g, W + (long)HH * VV, hg, UU, 0);
  }
  // Broadcast add + mask: 134,217,728 floats -> 33,554,432 float4 threads
  {
    const long n4 = (long)BB * TT * UU * (VV / 4);
    dim3 grid((unsigned)(n4 / 256));
    rnnt_joint_add<<<grid, 256, 0, stream>>>(hf, hg, bias, f_lens, g_lens, out);
  }
}